// ContinuousVariableQNN_33921651704032
// MI455X (gfx1250) — compile-verified
//
#include <hip/hip_runtime.h>
#include <math.h>

// ContinuousVariableQNN on MI455X (gfx1250, wave32, WMMA).
//
// Phase 1 (tiny): build the 2N x 2N symplectic S in d_ws, pack W[a,i] = S[i,2a]
//                 into a k-pair-interleaved layout for WMMA B fragments, and
//                 compute base[m] = diag(S S^T) pair-sums / 4 - 0.5.
// Phase 2 (big):  n[b,m] = base[m] + (nu_x^2 + nu_p^2)  where nu = inputs @ W
//                 (the displacement factor 2 and the 1/(2*hbar)=1/4 cancel:
//                  (2a.S)^2/4 = (a.S)^2), done via V_WMMA_F32_16X16X4_F32.
//
// Data-movement plan (HBM floor = 64 MB in + 64 MB out ~ 5.5 us @ 23.3 TB/s):
//   - W (128 KB) is read ONCE per wave into registers: each wave owns two
//     16-wide column tiles (64 B-fragments = 128 VGPRs).
//   - A streams from HBM exactly once; the 8 waves of a block read the same
//     rows, so redundancy is absorbed by WGP$/L2.
//   - Two independent accumulators per wave break the WMMA D->C serial chain;
//     the hot loop is purely global_load_b64 + 2x WMMA per k-step.
//   - mu is never materialized: fused shuffle epilogue writes n directly (NT).

typedef __attribute__((ext_vector_type(2))) float v2f;
typedef __attribute__((ext_vector_type(8))) float v8f;

#define NQ      128      // modes
#define TWO_N   256
#define NLAYERS 8
#define NBATCH  131072
#define ROWS_PER_BLOCK 256   // 16 row tiles of 16; grid = 512 blocks

// d_ws layout in floats:
//   [0,      32768)  WP: packed B operand. element (k,col) at ((k>>1)*256+col)*2 + (k&1)
//   [32768,  32896)  base[m] = 0.25*(dV[2m]+dV[2m+1]) - 0.5
//   [33024,  98560)  S, row-major S[r*256 + j]
#define WS_WP   0
#define WS_BASE 32768
#define WS_S    33024

// ---------------------------------------------------------------- phase 1 ----
__global__ __launch_bounds__(256) void cvqnn_build_S(
    const float* __restrict__ params, float* __restrict__ ws)
{
  __shared__ float sblk[NLAYERS][NQ][4];   // per-mode 2x2 block, 16 KB
  __shared__ float sdV[TWO_N];             // diag(S S^T) accumulators

  const int j = threadIdx.x;               // this thread owns column j of S
  float* S = ws + WS_S;

  // per-mode local block: R(th2) @ diag(e^-r, e^r) @ R(th1)
  for (int t = j; t < NLAYERS * NQ; t += 256) {
    const int l = t >> 7, n = t & (NQ - 1);
    const float th1 = params[l * 384 + n * 3 + 0];
    const float r   = params[l * 384 + n * 3 + 1];
    const float th2 = params[l * 384 + n * 3 + 2];
    const float c1 = cosf(th1), s1 = sinf(th1);
    const float c2 = cosf(th2), s2 = sinf(th2);
    const float em = expf(-r),  ep = expf(r);
    sblk[l][n][0] =  c2 * em * c1 - s2 * ep * s1;   // a
    sblk[l][n][1] = -c2 * em * s1 - s2 * ep * c1;   // b
    sblk[l][n][2] =  s2 * em * c1 + c2 * ep * s1;   // c
    sblk[l][n][3] = -s2 * em * s1 + c2 * ep * c1;   // d
  }
  sdV[j] = 0.0f;

  // S = I (column j). Row-major so simultaneous same-row access is coalesced.
  for (int r = 0; r < TWO_N; ++r) S[r * TWO_N + j] = (r == j) ? 1.0f : 0.0f;
  __syncthreads();   // for sblk / sdV; S itself is thread-private per column

  const float ct = 0.70710678118654752440f;  // cos(pi/4) = sin(pi/4)
  for (int l = 0; l < NLAYERS; ++l) {
    // D step: rows (2n,2n+1) <- blk[l][n] @ rows
    for (int n = 0; n < NQ; ++n) {
      const float x = S[(2 * n    ) * TWO_N + j];
      const float p = S[(2 * n + 1) * TWO_N + j];
      S[(2 * n    ) * TWO_N + j] = sblk[l][n][0] * x + sblk[l][n][1] * p;
      S[(2 * n + 1) * TWO_N + j] = sblk[l][n][2] * x + sblk[l][n][3] * p;
    }
    // C step: serial beamsplitter chain down the rows, carrying rows 2i+2,2i+3
    float v0 = S[0 * TWO_N + j], v1 = S[1 * TWO_N + j];
    for (int i = 0; i < NQ - 1; ++i) {
      const float v2 = S[(2 * i + 2) * TWO_N + j];
      const float v3 = S[(2 * i + 3) * TWO_N + j];
      const float n0 = ct * v0 - ct * v2;
      const float n1 = ct * v1 - ct * v3;
      const float n2 = ct * v0 + ct * v2;
      const float n3 = ct * v1 + ct * v3;
      S[(2 * i    ) * TWO_N + j] = n0;
      S[(2 * i + 1) * TWO_N + j] = n1;
      v0 = n2; v1 = n3;
    }
    S[(TWO_N - 2) * TWO_N + j] = v0;
    S[(TWO_N - 1) * TWO_N + j] = v1;
  }

  // Pack B operand: W[a,i] = S[i, 2a]; pairs of consecutive k interleaved so a
  // WMMA B fragment is one aligned float2 load. Only even columns contribute.
  if ((j & 1) == 0) {
    const int a = j >> 1;                              // k index 0..127
    float* wp = ws + WS_WP;
    const int b0 = (a >> 1) * 512 + (a & 1);           // ((a>>1)*256+i)*2+(a&1)
    for (int i = 0; i < TWO_N; ++i) wp[b0 + 2 * i] = S[i * TWO_N + j];
  }

  // dV[i] = sum_j S[i,j]^2 : rotate start row so lanes hit distinct LDS words
  for (int t = 0; t < TWO_N; ++t) {
    const int i = (j + t) & (TWO_N - 1);
    const float s = S[i * TWO_N + j];
    atomicAdd(&sdV[i], s * s);
  }
  __syncthreads();
  if (j < NQ) {
    float* base = ws + WS_BASE;
    base[j] = 0.25f * (sdV[2 * j] + sdV[2 * j + 1]) - 0.5f;
  }
}

// ---------------------------------------------------------------- phase 2 ----
// 512 blocks x 256 threads (8 waves). Wave w owns output columns
// [32w, 32w+32) (two 16-wide tiles) with B fragments register-resident, and
// loops over the block's 16 row-tiles streaming A from global.
__global__ __launch_bounds__(256) void cvqnn_gemm(
    const float* __restrict__ inputs, const float* __restrict__ ws,
    float* __restrict__ out)
{
  const int lane = threadIdx.x & 31;
  const int wave = threadIdx.x >> 5;
  const int m  = lane & 15;   // M (A) or N (B/C/D) index within the fragment
  const int kh = lane >> 4;   // which K-half of the 16x16x4 fragment

  const float* __restrict__ wp   = ws + WS_WP;
  const float* __restrict__ base = ws + WS_BASE;

  // ---- preload B fragments for this wave's two column tiles (read once) ----
  const int c0 = wave * 32;          // tile 0: cols c0..c0+15, tile 1: +16
  v2f bfrag0[32], bfrag1[32];
  {
    // element (k,col) lives at wp[ ((k>>1)*256 + col)*2 + (k&1) ];
    // fragment kb uses k = 4*kb + 2*kh (even), so float2 at kb*1024 + ...
    const float* b0 = wp + (size_t)(kh * 256 + c0 + m) * 2;
    const float* b1 = b0 + 32;       // +16 columns * 2
#pragma unroll
    for (int kb = 0; kb < 32; ++kb) {
      bfrag0[kb] = *(const v2f*)(b0 + (size_t)kb * 1024);
      bfrag1[kb] = *(const v2f*)(b1 + (size_t)kb * 1024);
    }
  }

  // per-lane epilogue constants (columns are fixed for this wave)
  const int mode0 = (c0 + m) >> 1;
  const int mode1 = (c0 + 16 + m) >> 1;
  const float bs0 = base[mode0];
  const float bs1 = base[mode1];

  // ---- loop over the block's row tiles, streaming A ----
  for (int rt = 0; rt < ROWS_PER_BLOCK / 16; ++rt) {
    const int rowbase = blockIdx.x * ROWS_PER_BLOCK + rt * 16;
    const float* arow = inputs + (size_t)(rowbase + m) * NQ + 2 * kh;

    v8f acc0 = {};
    v8f acc1 = {};
#pragma unroll
    for (int kb = 0; kb < 32; ++kb) {
      // A used UNSCALED: the 2x displacement and the 1/4 in n cancel exactly.
      const v2f a = __builtin_nontemporal_load((const v2f*)(arow + 4 * kb));
      acc0 = __builtin_amdgcn_wmma_f32_16x16x4_f32(
          false, a, false, bfrag0[kb], (short)0, acc0, false, false);
      acc1 = __builtin_amdgcn_wmma_f32_16x16x4_f32(
          false, a, false, bfrag1[kb], (short)0, acc1, false, false);
    }

    // epilogue: C/D layout -> row = rowbase + r + 8*kh, col = c0 + (lane&15).
    // lane^1 holds the partner quadrature column of the same row.
    // acc = mu/2, so n = base + acc_x^2 + acc_p^2 (no extra scaling needed).
    const int rowof = rowbase + (kh << 3);
#pragma unroll
    for (int r = 0; r < 8; ++r) {
      const float mu0  = acc0[r];
      const float mu0p = __shfl_xor(mu0, 1, 32);
      const float v0   = bs0 + (mu0 * mu0 + mu0p * mu0p);
      const float mu1  = acc1[r];
      const float mu1p = __shfl_xor(mu1, 1, 32);
      const float v1   = bs1 + (mu1 * mu1 + mu1p * mu1p);
      if ((lane & 1) == 0) {
        float* orow = out + (size_t)(rowof + r) * NQ;
        __builtin_nontemporal_store(v0, orow + mode0);
        __builtin_nontemporal_store(v1, orow + mode1);
      }
    }
  }
}

// ----------------------------------------------------------------- launch ----
extern "C" void kernel_launch(void* const* d_in, const int* in_sizes, int n_in,
                              void* d_out, int out_size, void* d_ws, size_t ws_size,
                              hipStream_t stream)
{
  const float* inputs = (const float*)d_in[0];  // [131072, 128] f32
  const float* params = (const float*)d_in[1];  // [8, 384] f32
  float* out = (float*)d_out;                   // [131072, 128] f32
  float* ws  = (float*)d_ws;                    // >= ~394 KB used

  cvqnn_build_S<<<1, 256, 0, stream>>>(params, ws);
  cvqnn_gemm<<<NBATCH / ROWS_PER_BLOCK, 256, 0, stream>>>(inputs, ws, out);
}